// Wavefront_SOS_2542620639464
// MI455X (gfx1250) — compile-verified
//
#include <hip/hip_runtime.h>
#include <hip/hip_bf16.h>
#include <math.h>

typedef __attribute__((ext_vector_type(2))) float v2f;
typedef __attribute__((ext_vector_type(8))) float v8f;

#define N_GRID   4096
#define N_POINTS 2048
#define N_INT    4096
#define THREADS  256

__global__ __launch_bounds__(THREADS) void wavefront_sos_kernel(
    const float* __restrict__ xp, const float* __restrict__ yp,
    const float* __restrict__ SOS, const float* __restrict__ x_vec,
    const float* __restrict__ y_vec, const float* __restrict__ thetas,
    const float* __restrict__ Rp, const float* __restrict__ v0p,
    float* __restrict__ out)
{
    const int ray  = blockIdx.x;
    const int tid  = threadIdx.x;
    const int lane = tid & 31;
    const int wid  = tid >> 5;

    const float x  = xp[0];
    const float y  = yp[0];
    const float R  = Rp[0];
    const float v0 = v0p[0];
    const float th = thetas[ray];

    // ---- per-ray geometry (computed redundantly by all threads; cheap) ----
    const float r   = sqrtf(x * x + y * y);
    const float phi = atan2f(x, y);                 // note arg order, as in reference
    const float s   = r * sinf(th - phi);
    const float c   = r * cosf(th - phi);
    const float disc = R * R - s * s;
    const float sq   = sqrtf(fmaxf(disc, 0.0f));
    const float l_in = sq + c;
    const bool  msk  = (cosf(phi - th) >= 0.0f) && (R >= fabsf(s));
    const float l_out = msk ? (sq + c) : 0.0f;
    const float l = (r < R) ? l_in : l_out;

    const float x0     = x_vec[0];
    const float yN     = y_vec[N_GRID - 1];
    const float inv_dx = 1.0f / (x_vec[1] - x_vec[0]);
    const float inv_dy = 1.0f / (y_vec[1] - y_vec[0]);
    const float dt     = 1.0f / (float)(N_INT - 1);

    const float sth = sinf(th);
    const float cth = cosf(th);
    const float lsin = l * sth;
    const float lcos = l * cth;

    // ---- gather + weighted sum: sample k handled by lane pattern tid + 256*it
    // (consecutive lanes -> consecutive samples -> cache-line locality in L2)
    float acc = 0.0f;
#pragma unroll
    for (int it = 0; it < N_INT / THREADS; ++it) {
        const int   k  = tid + it * THREADS;
        const float t  = (float)k * dt;
        const float xs = fmaf(-lsin, t, x);
        const float ys = fmaf(-lcos, t, y);
        int j = (int)rintf((xs - x0) * inv_dx);      // round-half-even == jnp.round
        int i = (int)rintf(-(ys - yN) * inv_dy);
        j = min(max(j, 0), N_GRID - 1);
        i = min(max(i, 0), N_GRID - 1);
        const float sos = SOS[i * N_GRID + j];
        const float f   = 1.0f - v0 / sos;
        const float w   = (k == 0 || k == N_INT - 1) ? 0.5f : 1.0f;  // trapezoid ends
        acc += w * f;
    }

    // ---- intra-wave (wave32) butterfly reduction ----
    for (int off = 16; off > 0; off >>= 1)
        acc += __shfl_xor(acc, off, 32);

    __shared__ float wsum[THREADS / 32];
    if (lane == 0) wsum[wid] = acc;
    __syncthreads();

    // ---- cross-wave reduction of 8 values via one V_WMMA_F32_16X16X4_F32 ----
    // A (16x4 f32, 2 VGPRs): lanes 0-15 hold M=0..15 with V0=K0,V1=K1;
    //                        lanes 16-31 hold M=0..15 with V0=K2,V1=K3.
    // Put wsum[0..3] in row 0, wsum[4..7] in row 1, zeros elsewhere.
    // B = all-ones (layout independent), C = 0.
    // Then D[m][n] = sum_k A[m][k]  =>  total = D[0][0] + D[1][0]
    // D layout: VGPR r, lane 0 holds (M=r, N=0)  =>  lane 0: d[0] + d[1].
    if (wid == 0) {                       // wave-uniform branch: EXEC all ones in wave 0
        const int m16 = lane & 15;
        const int hi  = lane >> 4;
        v2f a; a.x = 0.0f; a.y = 0.0f;
        if (m16 < 2) {
            const int base = m16 * 4 + hi * 2;
            a.x = wsum[base];
            a.y = wsum[base + 1];
        }
        v2f b; b.x = 1.0f; b.y = 1.0f;
        v8f cz = {};
        v8f d = __builtin_amdgcn_wmma_f32_16x16x4_f32(
            /*neg_a=*/false, a, /*neg_b=*/false, b,
            /*c_mod=*/(short)0, cz, /*reuse_a=*/false, /*reuse_b=*/false);
        if (lane == 0) {
            const float total = d[0] + d[1];
            out[ray]            = th;              // thetas passthrough (output 0)
            out[N_POINTS + ray] = l * dt * total;  // wf (output 1)
        }
    }
}

extern "C" void kernel_launch(void* const* d_in, const int* in_sizes, int n_in,
                              void* d_out, int out_size, void* d_ws, size_t ws_size,
                              hipStream_t stream) {
    // setup_inputs order: x, y, SOS, x_vec, y_vec, thetas, R_body, v0
    const float* x      = (const float*)d_in[0];
    const float* y      = (const float*)d_in[1];
    const float* SOS    = (const float*)d_in[2];
    const float* x_vec  = (const float*)d_in[3];
    const float* y_vec  = (const float*)d_in[4];
    const float* thetas = (const float*)d_in[5];
    const float* R_body = (const float*)d_in[6];
    const float* v0     = (const float*)d_in[7];
    float* out = (float*)d_out;

    wavefront_sos_kernel<<<dim3(N_POINTS), dim3(THREADS), 0, stream>>>(
        x, y, SOS, x_vec, y_vec, thetas, R_body, v0, out);
}